// QuantumFeedForwardBlock_65481071401074
// MI455X (gfx1250) — compile-verified
//
#include <hip/hip_runtime.h>

#define N_WIRES 8
#define N_OPS   20
#define ROWS    (4 * 4096)
#define EDIM    512
#define HDIM    64

typedef float v2f __attribute__((ext_vector_type(2)));
typedef float v8f __attribute__((ext_vector_type(8)));

struct OpsWS {
    int   kind[N_OPS];   // 0=rx 1=ry 2=rz 3=cnot
    int   wa[N_OPS];     // wire (or control)
    int   wb[N_OPS];     // target for cnot
    float gc[N_OPS];     // cos(theta/2)
    float gs[N_OPS];     // sin(theta/2)
    float fc[N_WIRES];   // final ry cos
    float fs[N_WIRES];   // final ry sin
};

// ---------------------------------------------------------------------------
// Prep kernel: reproduce numpy legacy RandomState(0) op sequence on device.
// init_genrand(0) + masked-rejection randint + Fisher-Yates permutation
// (numpy legacy choice(replace=False) == permutation(n)[:k]).
// ---------------------------------------------------------------------------
__global__ void qffb_prep(const float* __restrict__ rand_params,
                          const float* __restrict__ ry_theta,
                          OpsWS* __restrict__ ws) {
    __shared__ unsigned mt[624];
    if (threadIdx.x != 0) return;

    mt[0] = 0u;
    for (int i = 1; i < 624; ++i)
        mt[i] = 1812433253u * (mt[i - 1] ^ (mt[i - 1] >> 30)) + (unsigned)i;
    int mti = 624;

    auto next = [&]() -> unsigned {
        if (mti >= 624) {
            for (int i = 0; i < 624; ++i) {
                unsigned y = (mt[i] & 0x80000000u) | (mt[(i + 1) % 624] & 0x7fffffffu);
                mt[i] = mt[(i + 397) % 624] ^ (y >> 1) ^ ((y & 1u) ? 0x9908b0dfu : 0u);
            }
            mti = 0;
        }
        unsigned y = mt[mti++];
        y ^= y >> 11;
        y ^= (y << 7) & 0x9d2c5680u;
        y ^= (y << 15) & 0xefc60000u;
        y ^= y >> 18;
        return y;
    };
    auto interval = [&](unsigned mx) -> unsigned {
        unsigned mask = mx;
        mask |= mask >> 1;  mask |= mask >> 2;  mask |= mask >> 4;
        mask |= mask >> 8;  mask |= mask >> 16;
        unsigned v;
        do { v = next() & mask; } while (v > mx);
        return v;
    };

    for (int i = 0; i < N_OPS; ++i) {
        unsigned k = next() & 3u;
        if (k == 3u) {
            int p[8] = {0, 1, 2, 3, 4, 5, 6, 7};
            for (int j = 7; j >= 1; --j) {
                unsigned v = interval((unsigned)j);
                int tmp = p[j]; p[j] = p[v]; p[v] = tmp;
            }
            ws->kind[i] = 3; ws->wa[i] = p[0]; ws->wb[i] = p[1];
            ws->gc[i] = 0.f; ws->gs[i] = 0.f;
        } else {
            unsigned w = next() & 7u;
            ws->kind[i] = (int)k; ws->wa[i] = (int)w; ws->wb[i] = 0;
            float th = 0.5f * rand_params[i];
            ws->gc[i] = cosf(th); ws->gs[i] = sinf(th);
        }
    }
    for (int w = 0; w < N_WIRES; ++w) {
        float th = 0.5f * ry_theta[w];
        ws->fc[w] = cosf(th); ws->fs[w] = sinf(th);
    }
}

// Pair butterflies on slot (register) bits; all indices compile-time constant.
#define RY_PAIR(t0, t1) {                                   \
    float r0 = re[t0], i0 = im[t0], r1 = re[t1], i1 = im[t1]; \
    re[t0] = c * r0 - s * r1;  im[t0] = c * i0 - s * i1;      \
    re[t1] = s * r0 + c * r1;  im[t1] = s * i0 + c * i1; }

#define RX_PAIR(t0, t1) {                                   \
    float r0 = re[t0], i0 = im[t0], r1 = re[t1], i1 = im[t1]; \
    re[t0] = c * r0 + s * i1;  im[t0] = c * i0 - s * r1;      \
    re[t1] = c * r1 + s * i0;  im[t1] = c * i1 - s * r0; }

#define CSWAP_PAIR(t0, t1) {                                \
    int ctrl = (((((t0) << 5) | lane) >> bc) & 1);            \
    float r0 = re[t0], i0 = im[t0], r1 = re[t1], i1 = im[t1]; \
    re[t0] = ctrl ? r1 : r0;  im[t0] = ctrl ? i1 : i0;        \
    re[t1] = ctrl ? r0 : r1;  im[t1] = ctrl ? i0 : i1; }

#define SLOT_PAIRS(OPM, m)                                       \
    if ((m) == 1)      { OPM(0,1) OPM(2,3) OPM(4,5) OPM(6,7) }   \
    else if ((m) == 2) { OPM(0,2) OPM(1,3) OPM(4,6) OPM(5,7) }   \
    else               { OPM(0,4) OPM(1,5) OPM(2,6) OPM(3,7) }

// ---------------------------------------------------------------------------
// Main fused kernel: 512 threads = 16 waves; one wave simulates one row.
// Amplitude index i = (slot t << 5) | lane  (8 complex amps per lane).
// Wire w lives at bit (7-w): bits 5..7 are slot bits, bits 0..4 are lane bits.
// ---------------------------------------------------------------------------
__global__ __launch_bounds__(512)
void qffb_main(const float* __restrict__ x,
               const float* __restrict__ W1,
               const float* __restrict__ b1,
               const float* __restrict__ W2,
               const float* __restrict__ b2,
               const OpsWS* __restrict__ ws,
               float* __restrict__ out) {
    __shared__ float hS[16 * HDIM];

    const int lane = threadIdx.x & 31;
    const int wv   = threadIdx.x >> 5;
    const int row  = blockIdx.x * 16 + wv;

    // ---- features -> per-wire cos/sin ----
    float xv = 0.f;
    if (lane < N_WIRES) xv = x[(size_t)row * EDIM + lane];
    float cw[8], sw[8];
#pragma unroll
    for (int w = 0; w < 8; ++w) {
        float f = __shfl(xv, w, 32);
        cw[w] = cosf(0.5f * f);
        sw[w] = sinf(0.5f * f);
    }

    // ---- initial product state ----
    float lp = 1.f;
#pragma unroll
    for (int b = 0; b < 5; ++b)
        lp *= ((lane >> b) & 1) ? sw[7 - b] : cw[7 - b];
    float re[8], im[8];
#pragma unroll
    for (int t = 0; t < 8; ++t) {
        float sp = ((t >> 0) & 1) ? sw[2] : cw[2];   // bit 5 -> wire 2
        sp      *= ((t >> 1) & 1) ? sw[1] : cw[1];   // bit 6 -> wire 1
        sp      *= ((t >> 2) & 1) ? sw[0] : cw[0];   // bit 7 -> wire 0
        re[t] = lp * sp;
        im[t] = 0.f;
    }

    // ---- 20 fixed circuit ops ----
#pragma unroll 1
    for (int op = 0; op < N_OPS; ++op) {
        const int   kd = ws->kind[op];
        const int   a  = ws->wa[op];
        const float c  = ws->gc[op];
        const float s  = ws->gs[op];
        if (kd == 3) {                                // CNOT
            const int bc = 7 - a;
            const int bt = 7 - ws->wb[op];
            if (bt >= 5) {
                const int m = 1 << (bt - 5);
                SLOT_PAIRS(CSWAP_PAIR, m)
            } else {
                const int lm = 1 << bt;
#pragma unroll
                for (int t = 0; t < 8; ++t) {
                    int ctrl = (((t << 5) | lane) >> bc) & 1;
                    float pr = __shfl_xor(re[t], lm, 32);
                    float pi = __shfl_xor(im[t], lm, 32);
                    re[t] = ctrl ? pr : re[t];
                    im[t] = ctrl ? pi : im[t];
                }
            }
        } else if (kd == 2) {                         // RZ (pure phase)
            const int b = 7 - a;
#pragma unroll
            for (int t = 0; t < 8; ++t) {
                int bit = (((t << 5) | lane) >> b) & 1;
                float q = bit ? -s : s;
                float r = re[t], ii = im[t];
                re[t] = c * r + q * ii;
                im[t] = c * ii - q * r;
            }
        } else {                                      // RX / RY
            const int b = 7 - a;
            if (b >= 5) {
                const int m = 1 << (b - 5);
                if (kd == 0) { SLOT_PAIRS(RX_PAIR, m) }
                else         { SLOT_PAIRS(RY_PAIR, m) }
            } else {
                const int lm = 1 << b;
                if (kd == 0) {                        // RX: same formula both halves
#pragma unroll
                    for (int t = 0; t < 8; ++t) {
                        float pr = __shfl_xor(re[t], lm, 32);
                        float pi = __shfl_xor(im[t], lm, 32);
                        float r = re[t], ii = im[t];
                        re[t] = c * r + s * pi;
                        im[t] = c * ii - s * pr;
                    }
                } else {                              // RY
                    int bit = (lane >> b) & 1;
                    float q = bit ? s : -s;
#pragma unroll
                    for (int t = 0; t < 8; ++t) {
                        float pr = __shfl_xor(re[t], lm, 32);
                        float pi = __shfl_xor(im[t], lm, 32);
                        re[t] = c * re[t] + q * pr;
                        im[t] = c * im[t] + q * pi;
                    }
                }
            }
        }
    }

    // ---- trailing RY(ry_theta[w]) on every wire (compile-time placement) ----
    {
        float c, s;
        c = ws->fc[0]; s = ws->fs[0]; SLOT_PAIRS(RY_PAIR, 4)
        c = ws->fc[1]; s = ws->fs[1]; SLOT_PAIRS(RY_PAIR, 2)
        c = ws->fc[2]; s = ws->fs[2]; SLOT_PAIRS(RY_PAIR, 1)
#pragma unroll
        for (int w = 3; w < 8; ++w) {
            c = ws->fc[w]; s = ws->fs[w];
            const int b  = 7 - w;
            const int lm = 1 << b;
            int bit = (lane >> b) & 1;
            float q = bit ? s : -s;
#pragma unroll
            for (int t = 0; t < 8; ++t) {
                float pr = __shfl_xor(re[t], lm, 32);
                float pi = __shfl_xor(im[t], lm, 32);
                re[t] = c * re[t] + q * pr;
                im[t] = c * im[t] + q * pi;
            }
        }
    }

    // ---- probs -> Z expectations (8), wave butterfly reduction ----
    float p[8], P = 0.f;
#pragma unroll
    for (int t = 0; t < 8; ++t) { p[t] = re[t] * re[t] + im[t] * im[t]; P += p[t]; }
    float ev[8];
    ev[0] = ev[1] = ev[2] = 0.f;
#pragma unroll
    for (int t = 0; t < 8; ++t) {
        ev[0] += ((t >> 2) & 1) ? -p[t] : p[t];   // wire 0 -> bit 7 -> slot bit 2
        ev[1] += ((t >> 1) & 1) ? -p[t] : p[t];   // wire 1
        ev[2] += ((t >> 0) & 1) ? -p[t] : p[t];   // wire 2
    }
#pragma unroll
    for (int w = 3; w < 8; ++w)
        ev[w] = ((lane >> (7 - w)) & 1) ? -P : P;
#pragma unroll
    for (int off = 1; off < 32; off <<= 1) {
#pragma unroll
        for (int w = 0; w < 8; ++w) ev[w] += __shfl_xor(ev[w], off, 32);
    }

    // ---- layer 1: 8 -> 64, ReLU; each lane produces h[lane], h[lane+32] ----
    const float4* W1v = (const float4*)W1;
    float h0 = b1[lane], h1 = b1[lane + 32];
    {
        float4 a0 = W1v[lane * 2], a1 = W1v[lane * 2 + 1];
        h0 += ev[0] * a0.x + ev[1] * a0.y + ev[2] * a0.z + ev[3] * a0.w
            + ev[4] * a1.x + ev[5] * a1.y + ev[6] * a1.z + ev[7] * a1.w;
        float4 c0 = W1v[(lane + 32) * 2], c1 = W1v[(lane + 32) * 2 + 1];
        h1 += ev[0] * c0.x + ev[1] * c0.y + ev[2] * c0.z + ev[3] * c0.w
            + ev[4] * c1.x + ev[5] * c1.y + ev[6] * c1.z + ev[7] * c1.w;
    }
    hS[wv * HDIM + lane]      = fmaxf(h0, 0.f);
    hS[wv * HDIM + 32 + lane] = fmaxf(h1, 0.f);
    __syncthreads();

    // ---- layer 2: (16 x 64) @ (64 x 512) via V_WMMA_F32_16X16X4_F32 ----
    // Wave wv owns column tiles 2*wv and 2*wv+1 of this block's 16-row stripe.
    const int mrow  = lane & 15;
    const int khalf = (lane >> 4) << 1;     // 0 for lanes 0-15, 2 for 16-31

    v2f afr[16];
#pragma unroll
    for (int kk = 0; kk < 16; ++kk) {
        afr[kk].x = hS[mrow * HDIM + kk * 4 + khalf];
        afr[kk].y = hS[mrow * HDIM + kk * 4 + khalf + 1];
    }

#pragma unroll
    for (int nt = 0; nt < 2; ++nt) {
        const int col = (wv * 2 + nt) * 16 + mrow;
        float bv = b2[col];
        v8f acc = {bv, bv, bv, bv, bv, bv, bv, bv};
        const float* w2row = W2 + (size_t)col * HDIM;   // B[k][n] = W2[n][k]
#pragma unroll
        for (int kk = 0; kk < 16; ++kk) {
            v2f bfr;
            bfr.x = w2row[kk * 4 + khalf];
            bfr.y = w2row[kk * 4 + khalf + 1];
            acc = __builtin_amdgcn_wmma_f32_16x16x4_f32(
                false, afr[kk], false, bfr, (short)0, acc, false, false);
        }
        const int orow0 = blockIdx.x * 16 + ((lane >> 4) << 3);
        float* op = out + (size_t)orow0 * EDIM + col;
#pragma unroll
        for (int r = 0; r < 8; ++r) op[(size_t)r * EDIM] = acc[r];
    }
}

extern "C" void kernel_launch(void* const* d_in, const int* in_sizes, int n_in,
                              void* d_out, int out_size, void* d_ws, size_t ws_size,
                              hipStream_t stream) {
    const float* x           = (const float*)d_in[0];
    const float* ry_theta    = (const float*)d_in[1];
    const float* rand_params = (const float*)d_in[2];
    const float* W1          = (const float*)d_in[3];
    const float* b1          = (const float*)d_in[4];
    const float* W2          = (const float*)d_in[5];
    const float* b2          = (const float*)d_in[6];
    OpsWS* ws = (OpsWS*)d_ws;

    qffb_prep<<<1, 32, 0, stream>>>(rand_params, ry_theta, ws);
    qffb_main<<<ROWS / 16, 512, 0, stream>>>(x, W1, b1, W2, b2, ws, (float*)d_out);
}